// GraphAttention2_70050916598254
// MI455X (gfx1250) — compile-verified
//
#include <hip/hip_runtime.h>
#include <hip/hip_bf16.h>

typedef __attribute__((ext_vector_type(2))) float v2f;
typedef __attribute__((ext_vector_type(8))) float v8f;

#define NNODES 50000
#define NEDGES 800000
#define NH     2
#define NIN    128
#define NOUT   32
#define NTILES ((NNODES + 15) / 16)

// --------------------------------------------------------------------------
// Kernel A: build the padded WMMA B operand.
// Logical Vpad[K][n] (128 x 16):
//   Vpad[K][0] = 2 * sum_o W[0][K][o]   (head-0 "nsum" vector)
//   Vpad[K][1] = 2 * sum_o W[1][K][o]   (head-1 "nsum" vector)
//   Vpad[K][n>=2] = 0
// Stored pre-swizzled for the 16x16x4 f32 WMMA B layout so each lane can do a
// single 8-byte LDS load per K-step:
//   Vg[kb*64 + hi*32 + n*2 + j] = Vpad[4*kb + 2*hi + j][n]
// --------------------------------------------------------------------------
__global__ void build_b_kernel(const float* __restrict__ W, float* __restrict__ Vg) {
    int K = threadIdx.x;
    if (K >= NIN) return;
    float s0 = 0.f, s1 = 0.f;
    for (int o = 0; o < NOUT; ++o) {
        s0 += W[0 * NIN * NOUT + K * NOUT + o];
        s1 += W[1 * NIN * NOUT + K * NOUT + o];
    }
    s0 *= 2.0f;
    s1 *= 2.0f;
    int kb = K >> 2, r = K & 3, hi = r >> 1, j = r & 1;
    for (int n = 0; n < 16; ++n) {
        float v = (n == 0) ? s0 : (n == 1) ? s1 : 0.0f;
        Vg[kb * 64 + hi * 32 + n * 2 + j] = v;
    }
}

// --------------------------------------------------------------------------
// Kernel B: per-node reduced mat-vec via V_WMMA_F32_16X16X4_F32.
// One wave (32 lanes) per 16-node tile; 32 WMMA issues accumulate K=128.
// A layout (16x4 f32): lanes 0-15 hold M=0..15 K={0,1}, lanes 16-31 K={2,3}.
// B layout (4x16 f32): lanes 0-15 hold N=0..15 K={0,1}, lanes 16-31 K={2,3}.
// D layout (16x16 f32, 8 VGPRs): VGPR v -> M=v (lanes 0-15) / M=v+8 (16-31).
// --------------------------------------------------------------------------
__global__ __launch_bounds__(256) void node_gemm_kernel(const float* __restrict__ x,
                                                        const float* __restrict__ Vg,
                                                        float* __restrict__ nsum) {
    __shared__ float Vs[2048];
    for (int i = threadIdx.x; i < 2048; i += blockDim.x) Vs[i] = Vg[i];
    __syncthreads();

    int wavesPerBlock = blockDim.x >> 5;
    int gwave = blockIdx.x * wavesPerBlock + (threadIdx.x >> 5);
    if (gwave >= NTILES) return;           // wave-uniform: EXEC stays all-1s

    int lane = threadIdx.x & 31;
    int m    = lane & 15;                  // M for A-loads, N for B/D
    int hi   = lane >> 4;                  // K-half selector

    int node0 = gwave * 16;
    int nodeA = node0 + m;
    if (nodeA >= NNODES) nodeA = NNODES - 1;   // clamp (keeps EXEC full)
    const float* xrow = x + (size_t)nodeA * NIN;

    v8f c = {};
    #pragma unroll 4
    for (int kb = 0; kb < 32; ++kb) {
        v2f a, b;
        const float* ap = xrow + kb * 4 + hi * 2;      // 8B-aligned global load
        a.x = ap[0];
        a.y = ap[1];
        const float* bp = Vs + kb * 64 + hi * 32 + m * 2;  // 8B-aligned LDS load
        b.x = bp[0];
        b.y = bp[1];
        c = __builtin_amdgcn_wmma_f32_16x16x4_f32(false, a, false, b,
                                                  (short)0, c, false, false);
    }

    // Columns 0..1 of D hold nsum for heads 0,1.
    if (m < NH) {
        #pragma unroll
        for (int v = 0; v < 8; ++v) {
            int nodeO = node0 + v + hi * 8;
            if (nodeO < NNODES) nsum[m * NNODES + nodeO] = c[v];
        }
    }
}

// --------------------------------------------------------------------------
// Kernel C1: init per-node accumulators with the self-loop contribution
// (cnt = 1 edge, invd = 1/dist with dist = 1).
// --------------------------------------------------------------------------
__global__ void init_kernel(int* __restrict__ cnt, float* __restrict__ invd) {
    int n = blockIdx.x * blockDim.x + threadIdx.x;
    if (n < NNODES) {
        cnt[n]  = 1;
        invd[n] = 1.0f;
    }
}

// --------------------------------------------------------------------------
// Kernel C2: per-edge accumulation. Softmax collapsed exactly to 1/cnt, so we
// only need edge counts and sum of 1/dist per source node.
// --------------------------------------------------------------------------
__global__ void edge_acc_kernel(const float* __restrict__ pos,
                                const int* __restrict__ row,
                                const int* __restrict__ col,
                                int* __restrict__ cnt, float* __restrict__ invd) {
    int e = blockIdx.x * blockDim.x + threadIdx.x;
    if (e >= NEDGES) return;
    int r = row[e], cc = col[e];
    float dx = pos[cc * 3 + 0] - pos[r * 3 + 0];
    float dy = pos[cc * 3 + 1] - pos[r * 3 + 1];
    float dz = pos[cc * 3 + 2] - pos[r * 3 + 2];
    float d = sqrtf(dx * dx + dy * dy + dz * dz);
    d = fmaxf(d, 1e-6f);
    atomicAdd(&cnt[r], 1);
    atomicAdd(&invd[r], 1.0f / d);
}

// --------------------------------------------------------------------------
// Kernel D: node[h,n] = nsum[h,n] * invd[n] / cnt[n]
// --------------------------------------------------------------------------
__global__ void node_combine_kernel(const float* __restrict__ nsum,
                                    const int* __restrict__ cnt,
                                    const float* __restrict__ invd,
                                    float* __restrict__ node) {
    int i = blockIdx.x * blockDim.x + threadIdx.x;
    if (i >= NH * NNODES) return;
    int n = i % NNODES;
    node[i] = nsum[i] * invd[n] / (float)cnt[n];
}

// --------------------------------------------------------------------------
// Kernel E: out[h, e2] = node[h,row2[e2]] - node[h,col2[e2]]; self-loop
// entries (e2 >= E) are exactly zero.
// --------------------------------------------------------------------------
__global__ void output_kernel(const float* __restrict__ node,
                              const int* __restrict__ row,
                              const int* __restrict__ col,
                              float* __restrict__ out) {
    const int E2 = NEDGES + NNODES;
    int i = blockIdx.x * blockDim.x + threadIdx.x;
    if (i >= NH * E2) return;
    int h = i / E2;
    int e = i - h * E2;
    float v = 0.0f;
    if (e < NEDGES) {
        v = node[h * NNODES + row[e]] - node[h * NNODES + col[e]];
    }
    out[i] = v;
}

extern "C" void kernel_launch(void* const* d_in, const int* in_sizes, int n_in,
                              void* d_out, int out_size, void* d_ws, size_t ws_size,
                              hipStream_t stream) {
    const float* x   = (const float*)d_in[0];   // (N, 128)
    const float* pos = (const float*)d_in[1];   // (N, 3)
    const float* W   = (const float*)d_in[2];   // (2, 128, 32)
    // d_in[3] = attn: cancels exactly out of the result (softmax collapse)
    const int* row   = (const int*)d_in[4];     // (E,)
    const int* col   = (const int*)d_in[5];     // (E,)
    float* out = (float*)d_out;                 // (2, E+N)

    // Workspace carve-out (~1.2 MB total)
    float* wsf  = (float*)d_ws;
    float* Vg   = wsf;                          // 2048 floats (padded B operand)
    float* nsum = Vg + 2048;                    // 2*N floats
    float* invd = nsum + 2 * NNODES;            // N floats
    float* node = invd + NNODES;                // 2*N floats
    int*   cnt  = (int*)(node + 2 * NNODES);    // N ints

    build_b_kernel<<<1, 128, 0, stream>>>(W, Vg);

    int gemmBlocks = (NTILES + 7) / 8;          // 8 waves (tiles) per 256-thr block
    node_gemm_kernel<<<gemmBlocks, 256, 0, stream>>>(x, Vg, nsum);

    init_kernel<<<(NNODES + 255) / 256, 256, 0, stream>>>(cnt, invd);
    edge_acc_kernel<<<(NEDGES + 255) / 256, 256, 0, stream>>>(pos, row, col, cnt, invd);
    node_combine_kernel<<<(NH * NNODES + 255) / 256, 256, 0, stream>>>(nsum, cnt, invd, node);

    const int totalOut = NH * (NEDGES + NNODES);
    output_kernel<<<(totalOut + 255) / 256, 256, 0, stream>>>(node, row, col, out);
}